// RecurrentLayer_19241453486460
// MI455X (gfx1250) — compile-verified
//
#include <hip/hip_runtime.h>

// ---------------- problem constants ----------------
#define H        65
#define T        512
#define BATCH    256
#define NLAYER   10

// ---------------- tiling constants -----------------
#define KP       68          // K padded to 17*4 for f32 WMMA (16x16x4)
#define KSTEPS   17
#define NT       15          // N tiles: 3 gates * 5 tiles (gate padded 65->80)
#define GATE_NP  80
#define ROWS     16          // batch rows per workgroup (WMMA M)
#define WAVES    NT
#define NTHREADS (WAVES * 32)

typedef float v2f __attribute__((ext_vector_type(2)));
typedef float v8f __attribute__((ext_vector_type(8)));

static_assert(BATCH % ROWS == 0, "batch tiling");

// -------- async global->LDS prefetch (CDNA5), with sync fallback --------
#if defined(__gfx1250__) && __has_builtin(__builtin_amdgcn_global_load_async_to_lds_b32) && \
    __has_builtin(__builtin_amdgcn_s_wait_asynccnt)
#define USE_ASYNC 1
#else
#define USE_ASYNC 0
#endif

// Fetch one [ROWS][H] x-tile (global is contiguous: element e == sp[e]) into an
// LDS buffer laid out [ROWS][KP] (pad cols 65..67 pre-zeroed, never written).
__device__ __forceinline__
void x_tile_fetch(const float* __restrict__ sp, float* bufBase, int tid)
{
    for (int e = tid; e < ROWS * H; e += NTHREADS) {
        const int m = e / H, c = e % H;
#if USE_ASYNC
        __builtin_amdgcn_global_load_async_to_lds_b32(
            (__attribute__((address_space(1))) int*)(sp + e),
            (__attribute__((address_space(3))) int*)(bufBase + m * KP + c),
            0, 0);
#else
        bufBase[m * KP + c] = sp[e];
#endif
    }
}

__device__ __forceinline__ void x_tile_wait()
{
#if USE_ASYNC
    __builtin_amdgcn_s_wait_asynccnt(0);
#endif
}

__device__ __forceinline__ float fast_sigmoid(float x)
{
    // x -> -inf: __expf(-x)=inf -> 0 ;  x -> +inf: __expf(-x)=0 -> 1  (no NaN)
    return 1.0f / (1.0f + __expf(-x));
}

__device__ __forceinline__ float fast_tanh(float x)
{
    // tanh(x) = 1 - 2/(exp(2x)+1): x->+inf: 1 ; x->-inf: -1  (no NaN)
    const float e2 = __expf(2.0f * x);
    return 1.0f - 2.0f / (e2 + 1.0f);
}

__global__ __launch_bounds__(NTHREADS)
void gru_fused_wmma(const float* __restrict__ x_in,
                    const float* __restrict__ w_ih,
                    const float* __restrict__ w_hh,
                    const float* __restrict__ b_ih,
                    const float* __restrict__ b_hh,
                    float* __restrict__ ws0,     // ping buffer  [T][BATCH][H]
                    float* __restrict__ outbuf)  // pong buffer + final output
{
    // LDS staging (8/16B aligned for ds_load_b64 A-fragments)
    __shared__ __align__(16) float sh_x[2][ROWS][KP];       // double-buffered x tile
    __shared__ __align__(16) float sh_h[2][ROWS][KP];       // double-buffered hidden state
    __shared__ __align__(16) float sh_rz[ROWS][2 * GATE_NP];// r|z pre-activations
    __shared__ __align__(16) float sh_xn[ROWS][GATE_NP];    // xn pre-activation
    __shared__ __align__(16) float sh_hn[ROWS][GATE_NP];    // hn pre-activation

    const int tid  = threadIdx.x;
    const int wave = tid >> 5;
    const int lane = tid & 31;
    const int nl   = lane & 15;   // N-col within tile (B/C frag) == M-row (A frag)
    const int kh   = lane >> 4;   // K-half selector for A/B frags, M-half for C frag
    const int b0   = blockIdx.x * ROWS;

    const int  gate  = wave / 5;            // 0=reset, 1=update, 2=new (PyTorch order)
    const int  ct    = wave % 5;            // tile index within gate
    const int  col   = ct * 16 + nl;        // column within gate, 0..79 (valid < 65)
    const bool colOK = (col < H);
    const int  wrow  = gate * H + col;      // row of [3H][H] weight matrix

    // Zero both x-tile buffers once (pad cols 65..67 stay 0 forever; data cols
    // are rewritten every step). Barrier before any fetch touches them.
    for (int e = tid; e < 2 * ROWS * KP; e += NTHREADS) (&sh_x[0][0][0])[e] = 0.0f;
    __syncthreads();

    for (int layer = 0; layer < NLAYER; ++layer) {
        const float* src = (layer == 0) ? x_in : ((layer & 1) ? ws0 : outbuf);
        float*       dst = (layer & 1) ? outbuf : ws0;   // layer 9 (odd) -> outbuf
        const bool   relu = (layer == NLAYER - 1);

        // ---- kick off async prefetch of x[0] into buffer 0 ----
        x_tile_fetch(src + (size_t)b0 * H, &sh_x[0][0][0], tid);

        // ---- load this wave's weight B-fragments into registers (once/layer) ----
        const float* Wih = w_ih + layer * 3 * H * H;
        const float* Whh = w_hh + layer * 3 * H * H;
        v2f fih[KSTEPS], fhh[KSTEPS];
        for (int s = 0; s < KSTEPS; ++s) {
            const int k0 = 4 * s + 2 * kh;
            float a0 = 0.f, a1 = 0.f, c0 = 0.f, c1 = 0.f;
            if (colOK) {
                if (k0 < H)     { a0 = Wih[wrow * H + k0];     c0 = Whh[wrow * H + k0]; }
                if (k0 + 1 < H) { a1 = Wih[wrow * H + k0 + 1]; c1 = Whh[wrow * H + k0 + 1]; }
            }
            v2f fa; fa[0] = a0; fa[1] = a1; fih[s] = fa;
            v2f fc; fc[0] = c0; fc[1] = c1; fhh[s] = fc;
        }
        const float bih = colOK ? b_ih[layer * 3 * H + wrow] : 0.0f;
        const float bhh = colOK ? b_hh[layer * 3 * H + wrow] : 0.0f;

        // ---- h0 = 0 (both parity buffers; pad cols stay 0 forever) ----
        for (int e = tid; e < 2 * ROWS * KP; e += NTHREADS) (&sh_h[0][0][0])[e] = 0.0f;
        __syncthreads();

        for (int t = 0; t < T; ++t) {
            // x[t] tile was prefetched last iteration (or in the layer prologue)
            x_tile_wait();
            __syncthreads();   // x[t] visible; prev step's h/rz traffic ordered

            // prefetch x[t+1] into the other buffer while we compute
            if (t + 1 < T)
                x_tile_fetch(src + ((size_t)(t + 1) * BATCH + b0) * H,
                             &sh_x[(t + 1) & 1][0][0], tid);

            const float (*xb)[KP] = sh_x[t & 1];
            const float (*hb)[KP] = sh_h[t & 1];

            // ---- two independent 17-deep WMMA chains per wave ----
            v8f accx, acch;
            const float ix = (gate < 2) ? (bih + bhh) : bih;   // r/z: fold both biases
            const float ih = (gate < 2) ? 0.0f : bhh;          // n: keep hn separate
            for (int r = 0; r < 8; ++r) { accx[r] = ix; acch[r] = ih; }

            for (int s = 0; s < KSTEPS; ++s) {
                const int k0 = 4 * s + 2 * kh;
                v2f ax = *(const v2f*)&xb[nl][k0];
                v2f ah = *(const v2f*)&hb[nl][k0];
                accx = __builtin_amdgcn_wmma_f32_16x16x4_f32(
                    false, ax, false, fih[s], (short)0, accx, false, false);
                acch = __builtin_amdgcn_wmma_f32_16x16x4_f32(
                    false, ah, false, fhh[s], (short)0, acch, false, false);
            }

            // ---- scatter C fragments to LDS (row = r + 8*kh) ----
            {
                const int m0 = 8 * kh;
                if (gate == 0) {
                    for (int r = 0; r < 8; ++r) sh_rz[m0 + r][col] = accx[r] + acch[r];
                } else if (gate == 1) {
                    for (int r = 0; r < 8; ++r) sh_rz[m0 + r][GATE_NP + col] = accx[r] + acch[r];
                } else {
                    for (int r = 0; r < 8; ++r) {
                        sh_xn[m0 + r][col] = accx[r];
                        sh_hn[m0 + r][col] = acch[r];
                    }
                }
            }
            __syncthreads();

            // ---- gate nonlinearities + state update + stream out ----
            float*       dp  = dst + ((size_t)t * BATCH + b0) * H;
            float (*hnb)[KP] = sh_h[(t + 1) & 1];
            for (int e = tid; e < ROWS * H; e += NTHREADS) {
                const int m = e / H, c = e % H;
                const float r  = fast_sigmoid(sh_rz[m][c]);
                const float z  = fast_sigmoid(sh_rz[m][GATE_NP + c]);
                const float n  = fast_tanh(sh_xn[m][c] + r * sh_hn[m][c]);
                const float hv = (1.0f - z) * n + z * hb[m][c];
                hnb[m][c] = hv;
                dp[e] = relu ? fmaxf(hv, 0.0f) : hv;
            }
            // no trailing barrier: next iteration's wait+barrier orders everything
        }

        // Make this layer's global stores visible to next layer's async LDS loads
        // (async path may bypass WGP$), and order LDS reuse across layers.
        __threadfence();
        __syncthreads();
    }
}

extern "C" void kernel_launch(void* const* d_in, const int* in_sizes, int n_in,
                              void* d_out, int out_size, void* d_ws, size_t ws_size,
                              hipStream_t stream)
{
    const float* x    = (const float*)d_in[0];
    const float* w_ih = (const float*)d_in[1];
    const float* w_hh = (const float*)d_in[2];
    const float* b_ih = (const float*)d_in[3];
    const float* b_hh = (const float*)d_in[4];
    float* out = (float*)d_out;
    float* ws0 = (float*)d_ws;   // needs T*BATCH*H*4 = ~34 MB

    dim3 grid(BATCH / ROWS);     // 16 independent batch slabs
    dim3 block(NTHREADS);        // 480 threads = 15 wave32's (one per N-tile)
    gru_fused_wmma<<<grid, block, 0, stream>>>(x, w_ih, w_hh, b_ih, b_hh, ws0, out);
}